// Encoder_59021440582439
// MI455X (gfx1250) — compile-verified
//
#include <hip/hip_runtime.h>

typedef __attribute__((ext_vector_type(2))) float v2f;
typedef __attribute__((ext_vector_type(8))) float v8f;

// fp32 WMMA:  D(16x16) = A(16x4) * B(4x16) + C   -- immediates passed as literals (ICE-safe)
#define WMMA_K4(c_, a_, b_) \
  __builtin_amdgcn_wmma_f32_16x16x4_f32(false, (a_), false, (b_), (short)0, (c_), false, false)

__device__ __forceinline__ float silu_f(float x) { return x * (1.f / (1.f + __expf(-x))); }

// LDS row strides (padded so stride % 64 != 0 -> conflict-free 16-lane column reads)
#define XS   132   // 64 x 128 x-tile
#define XZS  516   // 64 x 512 xz-tile (xc cols 0..255, z cols 256..511)
#define DBS  52    // 64 x 48  dbc-tile (40 valid cols)
#define DTS  260   // 64 x 256 dt / y-scan tile
#define CS   260   // mix kernel 64 x 256
#define HS   516   // ffn hidden 64 x 512

// ---------------------------------------------------------------------------
// Kernel A: whole Mamba block, one workgroup per sequence (4096 sequences).
// Writes y (pre-mix) directly into d_out at its final concat slot.
// ---------------------------------------------------------------------------
__global__ void __launch_bounds__(256)
mamba_kernel(const float* __restrict__ x_node,
             const float* __restrict__ x_trace,
             const float* __restrict__ x_log,
             const float* __restrict__ mp_in,
             const float* __restrict__ mp_conv_w,
             const float* __restrict__ mp_conv_b,
             const float* __restrict__ mp_xproj,
             const float* __restrict__ mp_dt_w,
             const float* __restrict__ mp_dt_b,
             const float* __restrict__ mp_Alog,
             const float* __restrict__ mp_D,
             const float* __restrict__ mp_out,
             float* __restrict__ out)
{
  extern __shared__ float smem[];
  float* sX  = smem;                    // 64 x XS
  float* sXZ = sX  + 64 * XS;           // 64 x XZS
  float* sDB = sXZ + 64 * XZS;          // 64 x DBS
  float* sDT = sDB + 64 * DBS;          // 64 x DTS

  const int q = blockIdx.x;
  int strm, b, pos, nper, jslot;
  const float* xin;
  if (q < 1024)      { strm = 0; xin = x_node;  b = q >> 7; pos = q & 127; nper = 128; jslot = pos; }
  else if (q < 3072) { strm = 1; xin = x_trace; int t = q - 1024; b = t >> 8; pos = t & 255; nper = 256; jslot = 128 + pos; }
  else               { strm = 2; xin = x_log;   int t = q - 3072; b = t >> 7; pos = t & 127; nper = 128; jslot = 384 + pos; }

  const size_t in_base  = ((size_t)b * 64) * nper * 128 + (size_t)pos * 128;
  const size_t in_wstep = (size_t)nper * 128;

  const float* w_in   = mp_in     + (size_t)strm * 128 * 512;
  const float* conv_w = mp_conv_w + strm * 256 * 4;
  const float* conv_b = mp_conv_b + strm * 256;
  const float* w_xp   = mp_xproj  + strm * 256 * 40;
  const float* dt_w   = mp_dt_w   + strm * 8 * 256;
  const float* dt_b   = mp_dt_b   + strm * 256;
  const float* Alog   = mp_Alog   + strm * 256 * 16;
  const float* Dp     = mp_D      + strm * 256;
  const float* w_out  = mp_out    + (size_t)strm * 256 * 128;

  const int tid = threadIdx.x;
  const int lane = tid & 31, wid = tid >> 5;
  const int hl = lane >> 4, lm = lane & 15;
  const v8f vz = {};

  // ---- load x sequence tile (64 x 128) into LDS (coalesced)
  for (int idx = tid; idx < 64 * 128; idx += 256) {
    int w = idx >> 7, d = idx & 127;
    sX[w * XS + d] = xin[in_base + (size_t)w * in_wstep + d];
  }
  __syncthreads();

  // ---- GEMM1: xz(64x512) = x(64x128) @ w_in(128x512)
  // register-blocked 4(M) x 2(N) tiles per wave: 8 independent WMMA chains,
  // each B element fetched once per k-step (L2-resident weights read 1x/block).
  {
    const float* ab = sX + lm * XS + 2 * hl;
    for (int np = 0; np < 2; ++np) {
      const int n0a = (wid + np * 16) * 16;
      const int n0b = n0a + 128;
      const float* bpa = w_in + (2 * hl) * 512 + n0a + lm;
      const float* bpb = w_in + (2 * hl) * 512 + n0b + lm;
      v8f acc[8];
      #pragma unroll
      for (int u = 0; u < 8; ++u) acc[u] = vz;
      for (int k0 = 0; k0 < 128; k0 += 4) {
        v2f b0v; b0v.x = bpa[k0 * 512]; b0v.y = bpa[(k0 + 1) * 512];
        v2f b1v; b1v.x = bpb[k0 * 512]; b1v.y = bpb[(k0 + 1) * 512];
        v2f a[4];
        #pragma unroll
        for (int mt = 0; mt < 4; ++mt) {
          a[mt].x = ab[mt * 16 * XS + k0];
          a[mt].y = ab[mt * 16 * XS + k0 + 1];
        }
        #pragma unroll
        for (int mt = 0; mt < 4; ++mt) {
          acc[mt]     = WMMA_K4(acc[mt],     a[mt], b0v);
          acc[4 + mt] = WMMA_K4(acc[4 + mt], a[mt], b1v);
        }
      }
      #pragma unroll
      for (int mt = 0; mt < 4; ++mt)
        #pragma unroll
        for (int j = 0; j < 8; ++j) {
          sXZ[(mt * 16 + j + 8 * hl) * XZS + n0a + lm] = acc[mt][j];
          sXZ[(mt * 16 + j + 8 * hl) * XZS + n0b + lm] = acc[4 + mt][j];
        }
    }
  }
  __syncthreads();

  // ---- causal depthwise conv (K=4) + SiLU, in place; thread owns channel c
  {
    const int c = tid;
    const float cw0 = conv_w[c * 4 + 0], cw1 = conv_w[c * 4 + 1];
    const float cw2 = conv_w[c * 4 + 2], cw3 = conv_w[c * 4 + 3];
    const float cb  = conv_b[c];
    float xm3 = 0.f, xm2 = 0.f, xm1 = 0.f;
    for (int w = 0; w < 64; ++w) {
      float x0 = sXZ[w * XZS + c];
      float a = fmaf(cw0, xm3, fmaf(cw1, xm2, fmaf(cw2, xm1, fmaf(cw3, x0, cb))));
      sXZ[w * XZS + c] = silu_f(a);
      xm3 = xm2; xm2 = xm1; xm1 = x0;
    }
  }
  __syncthreads();

  // ---- GEMM2: dbc(64x40) = xc(64x256) @ w_xp(256x40)  (N padded to 48)
  for (int t = wid; t < 12; t += 8) {
    const int m0 = (t / 3) * 16, n0 = (t % 3) * 16;
    const int col = n0 + lm;
    const bool cok = col < 40;
    const float* arow = sXZ + (m0 + lm) * XZS + 2 * hl;
    const float* bp = w_xp + (2 * hl) * 40 + col;
    v8f acc = vz;
    #pragma unroll 4
    for (int k0 = 0; k0 < 256; k0 += 4) {
      v2f a;  a.x  = arow[k0];                 a.y  = arow[k0 + 1];
      v2f bv; bv.x = cok ? bp[k0 * 40] : 0.f;  bv.y = cok ? bp[(k0 + 1) * 40] : 0.f;
      acc = WMMA_K4(acc, a, bv);
    }
    #pragma unroll
    for (int j = 0; j < 8; ++j) sDB[(m0 + j + 8 * hl) * DBS + col] = acc[j];
  }
  __syncthreads();

  // ---- dt = softplus(dbc[:, :8] @ dt_w + dt_b)   (64 x 256), K=8 -> VALU
  {
    const int d = tid;
    const float bias = dt_b[d];
    for (int w = 0; w < 64; ++w) {
      const float* br = sDB + w * DBS;
      float s = bias;
      #pragma unroll
      for (int r = 0; r < 8; ++r) s = fmaf(br[r], dt_w[r * 256 + d], s);
      sDT[w * DTS + d] = (s > 20.f) ? s : log1pf(__expf(s));
    }
  }
  __syncthreads();

  // ---- selective scan over W=64; thread owns channel d, 16 SSM states in regs
  {
    const int d = tid;
    float Av[16], h[16];
    #pragma unroll
    for (int s = 0; s < 16; ++s) { Av[s] = -__expf(Alog[d * 16 + s]); h[s] = 0.f; }
    const float Dpv = Dp[d];
    for (int w = 0; w < 64; ++w) {
      const float dt = sDT[w * DTS + d];
      const float xt = sXZ[w * XZS + d];
      const float dtx = dt * xt;
      const float* Brow = sDB + w * DBS + 8;   // cols 8..23
      const float* Crow = sDB + w * DBS + 24;  // cols 24..39
      float y = 0.f;
      #pragma unroll
      for (int s = 0; s < 16; ++s) {
        float dA = __expf(dt * Av[s]);
        h[s] = fmaf(dA, h[s], dtx * Brow[s]);
        y = fmaf(h[s], Crow[s], y);
      }
      y = fmaf(xt, Dpv, y);
      const float z = sXZ[w * XZS + 256 + d];
      y *= silu_f(z);
      sDT[w * DTS + d] = y;                    // thread-exclusive in-place reuse
    }
  }
  __syncthreads();

  // ---- GEMM3: y(64x128) = yscan(64x256) @ w_out(256x128) -> d_out concat slot
  // 4(M) x 1(N) register block: one B fetch feeds 4 WMMA chains.
  {
    float* ybase = out + ((size_t)b * 64 * 512 + jslot) * 128;  // row stride 512*128
    const int n0 = wid * 16;
    const float* bp = w_out + (2 * hl) * 128 + n0 + lm;
    const float* ab = sDT + lm * DTS + 2 * hl;
    v8f acc[4];
    #pragma unroll
    for (int u = 0; u < 4; ++u) acc[u] = vz;
    for (int k0 = 0; k0 < 256; k0 += 4) {
      v2f bv; bv.x = bp[k0 * 128]; bv.y = bp[(k0 + 1) * 128];
      #pragma unroll
      for (int mt = 0; mt < 4; ++mt) {
        v2f a; a.x = ab[mt * 16 * DTS + k0]; a.y = ab[mt * 16 * DTS + k0 + 1];
        acc[mt] = WMMA_K4(acc[mt], a, bv);
      }
    }
    #pragma unroll
    for (int mt = 0; mt < 4; ++mt)
      #pragma unroll
      for (int j = 0; j < 8; ++j)
        ybase[(size_t)(mt * 16 + j + 8 * hl) * (512 * 128) + n0 + lm] = acc[mt][j];
  }
}

// ---------------------------------------------------------------------------
// Kernel B: node/log mix GEMM + residual + LayerNorm, in place on d_out.
// One workgroup per (b, n) pair; 1024 blocks.
// ---------------------------------------------------------------------------
__global__ void __launch_bounds__(256)
mix_ln_kernel(const float* __restrict__ x_node,
              const float* __restrict__ x_log,
              const float* __restrict__ mix_w,
              const float* __restrict__ mix_b,
              const float* __restrict__ an_g,
              const float* __restrict__ an_b,
              float* __restrict__ out)
{
  extern __shared__ float smem[];
  float* sCat = smem;             // 64 x CS
  float* sMix = smem + 64 * CS;   // 64 x CS

  const int q = blockIdx.x;
  const int b = q >> 7, n = q & 127;
  const int tid = threadIdx.x;
  const int lane = tid & 31, wid = tid >> 5;
  const int hl = lane >> 4, lm = lane & 15;
  const v8f vz = {};

  const float* ynp = out + ((size_t)b * 64 * 512 + n) * 128;
  const float* ylp = out + ((size_t)b * 64 * 512 + 384 + n) * 128;

  for (int idx = tid; idx < 64 * 256; idx += 256) {
    int w = idx >> 8, c = idx & 255;
    float v = (c < 128) ? ynp[(size_t)w * (512 * 128) + c]
                        : ylp[(size_t)w * (512 * 128) + (c - 128)];
    sCat[w * CS + c] = v;
  }
  __syncthreads();

  // mix = cat + silu(cat @ mix_w + mix_b)   (64x256 * 256x256), 4x2 reg block
  {
    const int n0a = wid * 16, n0b = n0a + 128;
    const float* bpa = mix_w + (2 * hl) * 256 + n0a + lm;
    const float* bpb = mix_w + (2 * hl) * 256 + n0b + lm;
    const float* ab = sCat + lm * CS + 2 * hl;
    v8f acc[8];
    #pragma unroll
    for (int u = 0; u < 8; ++u) acc[u] = vz;
    for (int k0 = 0; k0 < 256; k0 += 4) {
      v2f b0v; b0v.x = bpa[k0 * 256]; b0v.y = bpa[(k0 + 1) * 256];
      v2f b1v; b1v.x = bpb[k0 * 256]; b1v.y = bpb[(k0 + 1) * 256];
      v2f a[4];
      #pragma unroll
      for (int mt = 0; mt < 4; ++mt) {
        a[mt].x = ab[mt * 16 * CS + k0];
        a[mt].y = ab[mt * 16 * CS + k0 + 1];
      }
      #pragma unroll
      for (int mt = 0; mt < 4; ++mt) {
        acc[mt]     = WMMA_K4(acc[mt],     a[mt], b0v);
        acc[4 + mt] = WMMA_K4(acc[4 + mt], a[mt], b1v);
      }
    }
    const float mb0 = mix_b[n0a + lm], mb1 = mix_b[n0b + lm];
    #pragma unroll
    for (int mt = 0; mt < 4; ++mt)
      #pragma unroll
      for (int j = 0; j < 8; ++j) {
        int row = mt * 16 + j + 8 * hl;
        sMix[row * CS + n0a + lm] = sCat[row * CS + n0a + lm] + silu_f(acc[mt][j] + mb0);
        sMix[row * CS + n0b + lm] = sCat[row * CS + n0b + lm] + silu_f(acc[4 + mt][j] + mb1);
      }
  }
  __syncthreads();

  // residual + LayerNorm; token = (w, half); 2 threads per token, 64 ch each
  const int p = tid >> 1, sub = tid & 1;
  const int w = p >> 1, hsel = p & 1;
  const float* res = (hsel == 0 ? x_node : x_log)
                     + (((size_t)b * 64 + w) * 128 + n) * 128 + sub * 64;
  float* mrow = sMix + w * CS + hsel * 128 + sub * 64;
  float s1 = 0.f, s2 = 0.f;
  for (int k = 0; k < 64; ++k) {
    float v = mrow[k] + res[k];
    mrow[k] = v;
    s1 += v; s2 += v * v;
  }
  s1 += __shfl_xor(s1, 1);
  s2 += __shfl_xor(s2, 1);
  const float mean = s1 * (1.f / 128.f);
  const float var  = s2 * (1.f / 128.f) - mean * mean;
  const float rs   = rsqrtf(var + 1e-5f);
  const float* g  = an_g + (hsel ? 2 : 0) * 128 + sub * 64;
  const float* bb = an_b + (hsel ? 2 : 0) * 128 + sub * 64;
  const int jsl = (hsel == 0) ? n : (384 + n);
  float* orow = out + (((size_t)b * 64 + w) * 512 + jsl) * 128 + sub * 64;
  for (int k = 0; k < 64; ++k)
    orow[k] = (mrow[k] - mean) * rs * g[k] + bb[k];
}

// ---------------------------------------------------------------------------
// Kernel B2: trace residual + LayerNorm, in place. 2048 blocks of 128 threads.
// ---------------------------------------------------------------------------
__global__ void __launch_bounds__(128)
trace_ln_kernel(const float* __restrict__ x_trace,
                const float* __restrict__ an_g,
                const float* __restrict__ an_b,
                float* __restrict__ out)
{
  const int q = blockIdx.x;
  const int b = q >> 8, e = q & 255;
  const int tid = threadIdx.x;
  const int w = tid >> 1, sub = tid & 1;
  float* yrow = out + (((size_t)b * 64 + w) * 512 + 128 + e) * 128 + sub * 64;
  const float* res = x_trace + (((size_t)b * 64 + w) * 256 + e) * 128 + sub * 64;
  float vals[64];
  float s1 = 0.f, s2 = 0.f;
  for (int k = 0; k < 64; ++k) {
    float v = yrow[k] + res[k];
    vals[k] = v; s1 += v; s2 += v * v;
  }
  s1 += __shfl_xor(s1, 1);
  s2 += __shfl_xor(s2, 1);
  const float mean = s1 * (1.f / 128.f);
  const float var  = s2 * (1.f / 128.f) - mean * mean;
  const float rs   = rsqrtf(var + 1e-5f);
  const float* g  = an_g + 128 + sub * 64;
  const float* bb = an_b + 128 + sub * 64;
  for (int k = 0; k < 64; ++k)
    yrow[k] = (vals[k] - mean) * rs * g[k] + bb[k];
}

// ---------------------------------------------------------------------------
// Kernel C: FFN (128->512->128) + residual + LayerNorm, in place on d_out.
// 64 tokens per block; 4096 blocks. Stream boundaries (0/128/384) are 64-aligned.
// ---------------------------------------------------------------------------
__global__ void __launch_bounds__(256)
ffn_kernel(const float* __restrict__ ff1_w, const float* __restrict__ ff1_b,
           const float* __restrict__ ff2_w, const float* __restrict__ ff2_b,
           const float* __restrict__ fln_g, const float* __restrict__ fln_b,
           float* __restrict__ out)
{
  extern __shared__ float smem[];
  float* sXf = smem;              // 64 x 132
  float* sH  = sXf + 64 * 132;    // 64 x HS
  float* sY  = sH + 64 * HS;      // 64 x 132

  const int tid = threadIdx.x;
  const int lane = tid & 31, wid = tid >> 5;
  const int hl = lane >> 4, lm = lane & 15;
  const v8f vz = {};

  const size_t T0 = (size_t)blockIdx.x * 64;
  const int j0 = (int)(T0 & 511);
  const int i = (j0 < 128) ? 0 : (j0 < 384 ? 1 : 2);
  const float* w1 = ff1_w + (size_t)i * 128 * 512;
  const float* b1 = ff1_b + i * 512;
  const float* w2 = ff2_w + (size_t)i * 512 * 128;
  const float* b2 = ff2_b + i * 128;
  const float* g  = fln_g + i * 128;
  const float* bb = fln_b + i * 128;
  float* base = out + T0 * 128;

  for (int idx = tid; idx < 64 * 128; idx += 256) {
    int r = idx >> 7, c = idx & 127;
    sXf[r * 132 + c] = base[r * 128 + c];
  }
  __syncthreads();

  // H = leaky_relu(X @ w1 + b1)   (64x128 * 128x512), 4x2 reg block per pass
  {
    const float* ab = sXf + lm * 132 + 2 * hl;
    for (int np = 0; np < 2; ++np) {
      const int n0a = (wid + np * 16) * 16, n0b = n0a + 128;
      const float* bpa = w1 + (2 * hl) * 512 + n0a + lm;
      const float* bpb = w1 + (2 * hl) * 512 + n0b + lm;
      v8f acc[8];
      #pragma unroll
      for (int u = 0; u < 8; ++u) acc[u] = vz;
      for (int k0 = 0; k0 < 128; k0 += 4) {
        v2f b0v; b0v.x = bpa[k0 * 512]; b0v.y = bpa[(k0 + 1) * 512];
        v2f b1v; b1v.x = bpb[k0 * 512]; b1v.y = bpb[(k0 + 1) * 512];
        v2f a[4];
        #pragma unroll
        for (int mt = 0; mt < 4; ++mt) {
          a[mt].x = ab[mt * 16 * 132 + k0];
          a[mt].y = ab[mt * 16 * 132 + k0 + 1];
        }
        #pragma unroll
        for (int mt = 0; mt < 4; ++mt) {
          acc[mt]     = WMMA_K4(acc[mt],     a[mt], b0v);
          acc[4 + mt] = WMMA_K4(acc[4 + mt], a[mt], b1v);
        }
      }
      const float bv0 = b1[n0a + lm], bv1 = b1[n0b + lm];
      #pragma unroll
      for (int mt = 0; mt < 4; ++mt)
        #pragma unroll
        for (int j = 0; j < 8; ++j) {
          int row = mt * 16 + j + 8 * hl;
          float h0 = acc[mt][j] + bv0;     h0 = (h0 > 0.f) ? h0 : 0.01f * h0;
          float h1 = acc[4 + mt][j] + bv1; h1 = (h1 > 0.f) ? h1 : 0.01f * h1;
          sH[row * HS + n0a + lm] = h0;
          sH[row * HS + n0b + lm] = h1;
        }
    }
  }
  __syncthreads();

  // Y = H @ w2 + b2 + X   (64x512 * 512x128), 4x1 reg block
  {
    const int n0 = wid * 16;
    const float* bp = w2 + (2 * hl) * 128 + n0 + lm;
    const float* ab = sH + lm * HS + 2 * hl;
    v8f acc[4];
    #pragma unroll
    for (int u = 0; u < 4; ++u) acc[u] = vz;
    for (int k0 = 0; k0 < 512; k0 += 4) {
      v2f bv; bv.x = bp[k0 * 128]; bv.y = bp[(k0 + 1) * 128];
      #pragma unroll
      for (int mt = 0; mt < 4; ++mt) {
        v2f a; a.x = ab[mt * 16 * HS + k0]; a.y = ab[mt * 16 * HS + k0 + 1];
        acc[mt] = WMMA_K4(acc[mt], a, bv);
      }
    }
    const float bv2 = b2[n0 + lm];
    #pragma unroll
    for (int mt = 0; mt < 4; ++mt)
      #pragma unroll
      for (int j = 0; j < 8; ++j) {
        int r0 = mt * 16 + j + 8 * hl;
        sY[r0 * 132 + n0 + lm] = acc[mt][j] + bv2 + sXf[r0 * 132 + n0 + lm];
      }
  }
  __syncthreads();

  // LayerNorm per token: 4 threads/token, 32 channels each
  const int r = tid >> 2, sub = tid & 3;
  float s1 = 0.f, s2 = 0.f;
  for (int k = 0; k < 32; ++k) {
    float v = sY[r * 132 + sub * 32 + k];
    s1 += v; s2 += v * v;
  }
  s1 += __shfl_xor(s1, 1); s2 += __shfl_xor(s2, 1);
  s1 += __shfl_xor(s1, 2); s2 += __shfl_xor(s2, 2);
  const float mean = s1 * (1.f / 128.f);
  const float var  = s2 * (1.f / 128.f) - mean * mean;
  const float rs   = rsqrtf(var + 1e-5f);
  for (int k = 0; k < 32; ++k) {
    int c = sub * 32 + k;
    float v = sY[r * 132 + c];
    base[r * 128 + c] = (v - mean) * rs * g[c] + bb[c];
  }
}

// ---------------------------------------------------------------------------
extern "C" void kernel_launch(void* const* d_in, const int* in_sizes, int n_in,
                              void* d_out, int out_size, void* d_ws, size_t ws_size,
                              hipStream_t stream)
{
  (void)in_sizes; (void)n_in; (void)out_size; (void)d_ws; (void)ws_size;
  const float* x_node   = (const float*)d_in[0];
  const float* x_trace  = (const float*)d_in[1];
  const float* x_log    = (const float*)d_in[2];
  const float* mp_in    = (const float*)d_in[3];
  const float* mp_convw = (const float*)d_in[4];
  const float* mp_convb = (const float*)d_in[5];
  const float* mp_xproj = (const float*)d_in[6];
  const float* mp_dtw   = (const float*)d_in[7];
  const float* mp_dtb   = (const float*)d_in[8];
  const float* mp_Alog  = (const float*)d_in[9];
  const float* mp_D     = (const float*)d_in[10];
  const float* mp_out   = (const float*)d_in[11];
  const float* mix_w    = (const float*)d_in[12];
  const float* mix_b    = (const float*)d_in[13];
  const float* an_g     = (const float*)d_in[14];
  const float* an_b     = (const float*)d_in[15];
  const float* ff1_w    = (const float*)d_in[16];
  const float* ff1_b    = (const float*)d_in[17];
  const float* ff2_w    = (const float*)d_in[18];
  const float* ff2_b    = (const float*)d_in[19];
  const float* fln_g    = (const float*)d_in[20];
  const float* fln_b    = (const float*)d_in[21];
  float* out = (float*)d_out;

  const size_t smemA = (size_t)(64 * XS + 64 * XZS + 64 * DBS + 64 * DTS) * sizeof(float); // ~240 KB
  mamba_kernel<<<4096, 256, smemA, stream>>>(x_node, x_trace, x_log, mp_in, mp_convw,
      mp_convb, mp_xproj, mp_dtw, mp_dtb, mp_Alog, mp_D, mp_out, out);

  const size_t smemB = (size_t)(2 * 64 * CS) * sizeof(float);                              // ~130 KB
  mix_ln_kernel<<<1024, 256, smemB, stream>>>(x_node, x_log, mix_w, mix_b, an_g, an_b, out);

  trace_ln_kernel<<<2048, 128, 0, stream>>>(x_trace, an_g, an_b, out);

  const size_t smemC = (size_t)(64 * 132 + 64 * HS + 64 * 132) * sizeof(float);            // ~195 KB
  ffn_kernel<<<4096, 256, smemC, stream>>>(ff1_w, ff1_b, ff2_w, ff2_b, fln_g, fln_b, out);
}